// CosineSimilarity_23579370455461
// MI455X (gfx1250) — compile-verified
//
#include <hip/hip_runtime.h>
#include <hip/hip_bf16.h>

typedef __attribute__((ext_vector_type(2))) float v2f;
typedef __attribute__((ext_vector_type(4))) float v4f;
typedef __attribute__((ext_vector_type(8))) float v8f;

#define FEAT_D 128
#define NORM_EPS 1e-12f

// ---------------------------------------------------------------------------
// Kernel 1: per-row inverse L2 norm. One wave32 per row: each lane loads a
// float4 (32 lanes * 4 = 128 cols), squares, wave-reduces via shfl_xor.
// Writes inv[i] = 1 / max(||h_i||, eps) into workspace (N floats = 400 KB).
// ---------------------------------------------------------------------------
__global__ void cossim_invnorm_kernel(const float* __restrict__ h,
                                      float* __restrict__ inv, int N) {
  const int lane = threadIdx.x & 31;
  const int row  = blockIdx.x * (blockDim.x >> 5) + (threadIdx.x >> 5);
  if (row >= N) return;  // wave-uniform exit
  const float* rp = h + (size_t)row * FEAT_D + lane * 4;
  v4f v = *(const v4f*)rp;
  float s = v.x * v.x + v.y * v.y + v.z * v.z + v.w * v.w;
#pragma unroll
  for (int m = 16; m >= 1; m >>= 1) s += __shfl_xor(s, m, 32);
  if (lane == 0) {
    float n = sqrtf(s);
    inv[row] = 1.0f / fmaxf(n, NORM_EPS);
  }
}

// ---------------------------------------------------------------------------
// Kernel 2: SDDMM via V_WMMA_F32_16X16X4_F32. One wave handles 16 edges.
// A-frag: 16 src rows (16xK), B-frag: 16 dst rows as KxN. Both use the same
// per-lane float2 address pattern: row = edge (lane&15), col = k + 2*(lane>>4).
// After 32 WMMA steps the diagonal of the 16x16 accumulator is the 16 edge
// dot products; scale by the two inverse norms and store.
// ---------------------------------------------------------------------------
__global__ void cossim_sddmm_wmma_kernel(const float* __restrict__ h,
                                         const int* __restrict__ src,
                                         const int* __restrict__ dst,
                                         const float* __restrict__ inv,
                                         float* __restrict__ out, int E) {
  const int lane  = threadIdx.x & 31;
  const int group = blockIdx.x * (blockDim.x >> 5) + (threadIdx.x >> 5);
  const long long e0 = (long long)group * 16;
  if (e0 >= E) return;  // wave-uniform: EXEC stays all-ones below

  // Gather setup (clamped so partial tail groups stay fully convergent).
  const int r    = lane & 15;   // which edge of the group this lane serves
  const int half = lane >> 4;   // K sub-offset (0 or 2)
  long long eg = e0 + r;
  if (eg >= E) eg = (long long)E - 1;
  const int srow = src[eg];
  const int drow = dst[eg];
  const float* ap = h + (size_t)srow * FEAT_D + 2 * half;  // A: src rows
  const float* bp = h + (size_t)drow * FEAT_D + 2 * half;  // B: dst cols

  v8f acc = {};
#pragma unroll
  for (int k = 0; k < FEAT_D; k += 4) {
    v2f a = *(const v2f*)(ap + k);
    v2f b = *(const v2f*)(bp + k);
    // D = A(16x4) x B(4x16) + C, full f32 matrix pipe
    acc = __builtin_amdgcn_wmma_f32_16x16x4_f32(
        /*neg_a=*/false, a, /*neg_b=*/false, b,
        /*c_mod=*/(short)0, acc, /*reuse_a=*/false, /*reuse_b=*/false);
  }

  // Diagonal extraction: D[j][j] sits in (lane j, vgpr j) for j<8 and
  // (lane j+16, vgpr j-8) for j>=8  => active lanes 0..7 and 24..31.
  const bool active = (lane < 8) || (lane >= 24);
  const int  vidx   = lane & 7;
  const int  j      = (lane < 8) ? lane : (lane - 16);
  float diag = acc[0];
#pragma unroll
  for (int i = 1; i < 8; ++i) diag = (vidx == i) ? acc[i] : diag;

  const long long e = e0 + j;
  if (active && e < E) {
    const int sj = src[e];
    const int dj = dst[e];
    out[e] = diag * inv[sj] * inv[dj];
  }
}

extern "C" void kernel_launch(void* const* d_in, const int* in_sizes, int n_in,
                              void* d_out, int out_size, void* d_ws, size_t ws_size,
                              hipStream_t stream) {
  const float* h   = (const float*)d_in[0];
  const int*   src = (const int*)d_in[1];
  const int*   dst = (const int*)d_in[2];
  float*       out = (float*)d_out;
  float*       inv = (float*)d_ws;  // N floats of scratch

  const int N = in_sizes[0] / FEAT_D;
  const int E = in_sizes[1];

  // Kernel 1: inverse norms (8 rows per 256-thread block).
  {
    const int rows_per_block = 256 / 32;
    dim3 block(256), grid((N + rows_per_block - 1) / rows_per_block);
    cossim_invnorm_kernel<<<grid, block, 0, stream>>>(h, inv, N);
  }
  // Kernel 2: WMMA SDDMM (8 edge-groups of 16 per 256-thread block).
  {
    const int groups = (E + 15) / 16;
    const int gpb = 256 / 32;
    dim3 block(256), grid((groups + gpb - 1) / gpb);
    cossim_sddmm_wmma_kernel<<<grid, block, 0, stream>>>(h, src, dst, inv, out, E);
  }
}